// NCESoftmaxLoss_sam_1477468750346
// MI455X (gfx1250) — compile-verified
//
#include <hip/hip_runtime.h>
#include <hip/hip_bf16.h>
#include <stdint.h>

// Problem constants (from reference): x is [BSZ, NCLS] f32, NCLS = 32769.
#define NCLS        32769
#define ROW_THREADS 512              // 16 wave32 per block
#define ROW_WAVES   (ROW_THREADS / 32)
#define CHUNK       2048             // elements per wave (two TDM descriptors)
#define HCHUNK      1024             // per-descriptor elements
#define SEL_THREADS 1024

typedef __attribute__((ext_vector_type(2))) float        v2f;
typedef __attribute__((ext_vector_type(8))) float        v8f;
typedef __attribute__((ext_vector_type(4))) unsigned int v4u;
typedef __attribute__((ext_vector_type(8))) int          v8i;
typedef __attribute__((ext_vector_type(4))) int          v4i;

#if __has_builtin(__builtin_amdgcn_exp2f)
#define FAST_EXP2(x) __builtin_amdgcn_exp2f(x)   // raw v_exp_f32; args <= 0,
#else                                            // subnormal flush is harmless
#define FAST_EXP2(x) exp2f(x)
#endif

// ---------------------------------------------------------------------------
// One TDM descriptor: copy `nelem` f32 from global `gsrc` to LDS byte address
// `lds_byte_addr`. Degenerate-2D tensor, data_size = 4 bytes. D# layout per
// CDNA5 ISA §8. Groups 2/3/4 zero (dims unused). 6-arg builtin form.
// ---------------------------------------------------------------------------
__device__ __forceinline__ void tdm_load_chunk(const float* gsrc,
                                               unsigned lds_byte_addr,
                                               unsigned nelem) {
  uint64_t ga = (uint64_t)(uintptr_t)gsrc;
  v4u g0;
  g0.x = 1u;                                            // count=1, user mode
  g0.y = lds_byte_addr;                                 // lds_addr [63:32]
  g0.z = (unsigned)(ga & 0xFFFFFFFFu);                  // global_addr lo
  g0.w = (unsigned)((ga >> 32) & 0x01FFFFFFu) | (2u << 30); // addr hi | type=2
  v8i g1;
  g1[0] = (int)(2u << 16);                              // data_size=2 -> 4B
  g1[1] = (int)((nelem & 0xFFFFu) << 16);               // tensor_dim0[15:0]
  g1[2] = (int)(((nelem >> 16) & 0xFFFFu) | (1u << 16));// dim0 hi | tensor_dim1=1
  g1[3] = (int)((nelem & 0xFFFFu) << 16);               // tile_dim0 (<=65535)
  g1[4] = 1;                                            // tile_dim1=1
  g1[5] = (int)nelem;                                   // tensor_dim0_stride lo
  g1[6] = 0;
  g1[7] = 0;
  v4i z4 = {0, 0, 0, 0};
  v8i z8 = {0, 0, 0, 0, 0, 0, 0, 0};
  __builtin_amdgcn_tensor_load_to_lds(g0, g1, z4, z4, z8, 0);
}

// ---------------------------------------------------------------------------
// Kernel A: one block per row. Each wave issues TWO TDM descriptors (front and
// back half of its 2048-elem chunk), then pipelines: wait tensorcnt<=1 ->
// max-reduce over the first halves while the second halves stream in; wait
// tensorcnt==0 -> finish the max. Pass 2 sums exp2(t - M) through two
// independent V_WMMA_F32_16X16X4_F32 accumulators with all-ones A (sum(D) =
// 16 * sum(B), independent of B's lane layout).
// loss[row] = (M + log2(S)) * ln2 - x[row][0]
// ---------------------------------------------------------------------------
__global__ __launch_bounds__(ROW_THREADS)
void row_lse_kernel(const float* __restrict__ x, float* __restrict__ loss) {
  extern __shared__ float lds[];            // [32772] row (+pad) | [512] red
  float* red = lds + 32772;                 // 16B-aligned scratch
  const int  row = blockIdx.x;
  const int  tid = threadIdx.x;
  const float* xrow = x + (size_t)row * NCLS;

  // --- async-tensor stage: two descriptors per wave ---
  const int wave = __builtin_amdgcn_readfirstlane((int)(threadIdx.x >> 5));
  unsigned base1 = (unsigned)wave * CHUNK;
  unsigned base2 = base1 + HCHUNK;
  unsigned len2  = (wave == ROW_WAVES - 1) ? (unsigned)(NCLS - base2) : HCHUNK;
  unsigned lds_base = (unsigned)(uintptr_t)lds;   // low 32 bits = LDS offset
  tdm_load_chunk(xrow + base1, lds_base + base1 * 4u, HCHUNK);
  tdm_load_chunk(xrow + base2, lds_base + base2 * 4u, len2);

  const float LOG2E = 1.4426950408889634f;
  const float LN2   = 0.6931471805599453f;
  const float4* lds4 = (const float4*)lds;
  float xmax = -3.4e38f;

  // first descriptors done (per-wave in-order completion), sync block
  __builtin_amdgcn_s_wait_tensorcnt(1);
  __syncthreads();
  // half-1 regions: float4 idx in [w*512, w*512+256) for each wave w
  for (int v = tid; v < 4096; v += ROW_THREADS) {         // 8 exact iters
    int idx4 = ((v >> 8) << 9) + (v & 255);
    float4 q = lds4[idx4];
    xmax = fmaxf(xmax, fmaxf(fmaxf(q.x, q.y), fmaxf(q.z, q.w)));
  }

  // second descriptors done, sync block
  __builtin_amdgcn_s_wait_tensorcnt(0);
  __syncthreads();
  // half-2 regions: float4 idx in [w*512+256, w*512+512)
  for (int v = tid; v < 4096; v += ROW_THREADS) {         // 8 exact iters
    int idx4 = ((v >> 8) << 9) + 256 + (v & 255);
    float4 q = lds4[idx4];
    xmax = fmaxf(xmax, fmaxf(fmaxf(q.x, q.y), fmaxf(q.z, q.w)));
  }
  if (tid == 0) xmax = fmaxf(xmax, lds[NCLS - 1]);        // tail element

  red[tid] = xmax * LOG2E;
  __syncthreads();
  for (int s = ROW_THREADS / 2; s > 0; s >>= 1) {
    if (tid < s) red[tid] = fmaxf(red[tid], red[tid + s]);
    __syncthreads();
  }
  const float M = red[0];                                 // block-uniform
  __syncthreads();                                        // before reusing red

  // --- pass 2: sum exp2(x*log2e - M), two independent WMMA chains ---
  v2f aones; aones.x = 1.0f; aones.y = 1.0f;              // A = 16x4 ones
  v8f c0 = {0.f, 0.f, 0.f, 0.f, 0.f, 0.f, 0.f, 0.f};
  v8f c1 = {0.f, 0.f, 0.f, 0.f, 0.f, 0.f, 0.f, 0.f};
  for (int v = tid; v < (NCLS / 4); v += ROW_THREADS) {   // 8192/512 exact
    float4 q = lds4[v];
    v2f b0, b1;
    b0.x = FAST_EXP2(fmaf(q.x, LOG2E, -M));
    b0.y = FAST_EXP2(fmaf(q.y, LOG2E, -M));
    b1.x = FAST_EXP2(fmaf(q.z, LOG2E, -M));
    b1.y = FAST_EXP2(fmaf(q.w, LOG2E, -M));
    c0 = __builtin_amdgcn_wmma_f32_16x16x4_f32(false, aones, false, b0,
                                               (short)0, c0, false, false);
    c1 = __builtin_amdgcn_wmma_f32_16x16x4_f32(false, aones, false, b1,
                                               (short)0, c1, false, false);
  }
  float partial = (c0[0] + c0[1]) + (c0[2] + c0[3]) + (c0[4] + c0[5]) +
                  (c0[6] + c0[7]) + (c1[0] + c1[1]) + (c1[2] + c1[3]) +
                  (c1[4] + c1[5]) + (c1[6] + c1[7]);
  red[tid] = partial;
  __syncthreads();
  for (int s = ROW_THREADS / 2; s > 0; s >>= 1) {
    if (tid < s) red[tid] += red[tid + s];
    __syncthreads();
  }
  if (tid == 0) {
    float S = red[0] * (1.0f / 16.0f);                    // undo 16-row dup
    S += FAST_EXP2(fmaf(lds[NCLS - 1], LOG2E, -M));       // tail element
    float lse = (M + log2f(S)) * LN2;
    loss[row] = lse - lds[0];
  }
}

// ---------------------------------------------------------------------------
// Kernel B: single block. Bitonic-sort the 4096 losses in LDS, pick the
// quantile threshold p = sorted[floor(pp*(n-1))], masked mean of loss > p.
// ---------------------------------------------------------------------------
__global__ __launch_bounds__(SEL_THREADS)
void select_mean_kernel(const float* __restrict__ loss,
                        const float* __restrict__ pp,
                        float* __restrict__ out, int n) {
  __shared__ float s[4096];
  __shared__ float rs[SEL_THREADS];
  __shared__ int   rc[SEL_THREADS];
  const int tid = threadIdx.x;

  for (int i = tid; i < n; i += SEL_THREADS) s[i] = loss[i];
  __syncthreads();

  for (int k = 2; k <= n; k <<= 1) {
    for (int j = k >> 1; j > 0; j >>= 1) {
      for (int i = tid; i < n; i += SEL_THREADS) {
        int ixj = i ^ j;
        if (ixj > i) {
          float a = s[i], b = s[ixj];
          bool up = ((i & k) == 0);
          if ((a > b) == up) { s[i] = b; s[ixj] = a; }
        }
      }
      __syncthreads();
    }
  }

  int   idx = (int)floorf(pp[0] * (float)(n - 1));
  float p   = s[idx];

  float ls = 0.f; int lc = 0;
  for (int i = tid; i < n; i += SEL_THREADS) {
    float v = s[i];
    if (v > p) { ls += v; lc++; }
  }
  rs[tid] = ls; rc[tid] = lc;
  __syncthreads();
  for (int st = SEL_THREADS / 2; st > 0; st >>= 1) {
    if (tid < st) { rs[tid] += rs[tid + st]; rc[tid] += rc[tid + st]; }
    __syncthreads();
  }
  if (tid == 0) out[0] = rs[0] / (float)rc[0];
}

// ---------------------------------------------------------------------------
extern "C" void kernel_launch(void* const* d_in, const int* in_sizes, int n_in,
                              void* d_out, int out_size, void* d_ws,
                              size_t ws_size, hipStream_t stream) {
  const float* x  = (const float*)d_in[0];   // [bsz, NCLS] f32
  const float* pp = (const float*)d_in[1];   // scalar f32
  float* out  = (float*)d_out;               // scalar f32
  float* loss = (float*)d_ws;                // bsz f32 scratch

  const int bsz = in_sizes[0] / NCLS;        // 4096
  const size_t shmem = (size_t)(32772 + ROW_THREADS) * sizeof(float); // ~133KB

  row_lse_kernel<<<bsz, ROW_THREADS, shmem, stream>>>(x, loss);
  select_mean_kernel<<<1, SEL_THREADS, 0, stream>>>(loss, pp, out, bsz);
}